// CorotationalBeam2D_12146167513831
// MI455X (gfx1250) — compile-verified
//
#include <hip/hip_runtime.h>
#include <hip/hip_bf16.h>

// ---------------------------------------------------------------------------
// CorotationalBeam2D: bandwidth-bound gather/compute/scatter FE assembly.
// 4M elements, 1M nodes. Node data (24 MB) and nodal_forces (12 MB) are
// L2-resident on MI455X (192 MB L2); the 416 MB of per-element outputs are
// streamed once with non-temporal stores so they bypass-rinse out of L2 and
// leave the gathered node arrays hot. Scatter uses native global_atomic_add_f32.
// ---------------------------------------------------------------------------

__global__ __launch_bounds__(256) void beam2d_zero_kernel(float* __restrict__ p,
                                                          long long n) {
  long long i = (long long)blockIdx.x * blockDim.x + threadIdx.x;
  if (i < n) p[i] = 0.0f;
}

__global__ __launch_bounds__(256) void beam2d_elem_kernel(
    const float* __restrict__ pred_disp,   // (N,3)
    const int*   __restrict__ conn,        // (E,2) int32 pairs
    const float* __restrict__ coords,      // (N,3)
    const float* __restrict__ prop_E,      // (E,)
    const float* __restrict__ prop_A,      // (E,)
    const float* __restrict__ prop_I22,    // (E,)
    float* __restrict__ nodal_forces,      // (N,3) pre-zeroed, atomic accumulate
    float* __restrict__ f_local,           // (E,6)
    float* __restrict__ d_local,           // (E,6)
    float* __restrict__ F_A,               // (E,3)
    float* __restrict__ F_B,               // (E,3)
    float* __restrict__ N_e,               // (E,)
    float* __restrict__ M_mid,             // (E,)
    float* __restrict__ V_e,               // (E,)
    float* __restrict__ M1_e,              // (E,)
    float* __restrict__ M2_e,              // (E,)
    float* __restrict__ l0_out,            // (E,)
    float* __restrict__ c_out,             // (E,)
    float* __restrict__ s_out,             // (E,)
    int n_elems)
{
  int e = blockIdx.x * blockDim.x + threadIdx.x;
  if (e >= n_elems) return;

  // Prefetch the streamed connectivity ahead (gfx1250 global_prefetch_b8;
  // speculative — OOB translation failures are silently dropped).
  __builtin_prefetch(conn + 2 * e + 16384, 0, 0);

  // --- gather ---------------------------------------------------------------
  const int2 ab = reinterpret_cast<const int2*>(conn)[e];   // global_load_b64
  const int nA = ab.x;
  const int nB = ab.y;

  const float xA = coords[3 * nA + 0];
  const float zA = coords[3 * nA + 2];
  const float xB = coords[3 * nB + 0];
  const float zB = coords[3 * nB + 2];

  const float uxA = pred_disp[3 * nA + 0];
  const float uzA = pred_disp[3 * nA + 1];
  const float thA = pred_disp[3 * nA + 2];
  const float uxB = pred_disp[3 * nB + 0];
  const float uzB = pred_disp[3 * nB + 1];
  const float thB = pred_disp[3 * nB + 2];

  const float E_  = prop_E[e];
  const float A_  = prop_A[e];
  const float I22 = prop_I22[e];

  // --- element geometry / local frame --------------------------------------
  const float dx0 = xB - xA;
  const float dz0 = zB - zA;
  const float l0  = sqrtf(dx0 * dx0 + dz0 * dz0);
  const float c   = dx0 / l0;
  const float s   = dz0 / l0;

  const float EA = E_ * A_;
  const float EI = E_ * I22;

  const float ua = c * uxA + s * uzA;
  const float wa = -s * uxA + c * uzA;
  const float ub = c * uxB + s * uzB;
  const float wb = -s * uxB + c * uzB;

  const float L  = l0;
  const float L2 = L * L;
  const float L3 = L2 * L;

  const float EA_L = EA / L;
  const float k12  = 12.0f * EI / L3;
  const float k6   = 6.0f * EI / L2;
  const float EI_L = EI / L;

  const float f0 = EA_L * (ua - ub);
  const float f1 = k12 * (wa - wb) + k6 * (thA + thB);
  const float f2 = k6 * (wa - wb) + EI_L * (4.0f * thA + 2.0f * thB);
  const float f3 = EA_L * (ub - ua);
  const float f4 = k12 * (wb - wa) - k6 * (thA + thB);
  const float f5 = k6 * (wa - wb) + EI_L * (2.0f * thA + 4.0f * thB);

  const float FAx = c * f0 - s * f1;
  const float FAz = s * f0 + c * f1;
  const float FBx = c * f3 - s * f4;
  const float FBz = s * f3 + c * f4;

  // --- streamed per-element outputs: non-temporal (write-once, 416 MB) -----
  {
    float* fl = f_local + 6ll * e;
    __builtin_nontemporal_store(f0, fl + 0);
    __builtin_nontemporal_store(f1, fl + 1);
    __builtin_nontemporal_store(f2, fl + 2);
    __builtin_nontemporal_store(f3, fl + 3);
    __builtin_nontemporal_store(f4, fl + 4);
    __builtin_nontemporal_store(f5, fl + 5);
  }
  {
    float* dl = d_local + 6ll * e;
    __builtin_nontemporal_store(ua,  dl + 0);
    __builtin_nontemporal_store(wa,  dl + 1);
    __builtin_nontemporal_store(thA, dl + 2);
    __builtin_nontemporal_store(ub,  dl + 3);
    __builtin_nontemporal_store(wb,  dl + 4);
    __builtin_nontemporal_store(thB, dl + 5);
  }
  {
    float* fa = F_A + 3ll * e;
    __builtin_nontemporal_store(FAx, fa + 0);
    __builtin_nontemporal_store(FAz, fa + 1);
    __builtin_nontemporal_store(f2,  fa + 2);
    float* fb = F_B + 3ll * e;
    __builtin_nontemporal_store(FBx, fb + 0);
    __builtin_nontemporal_store(FBz, fb + 1);
    __builtin_nontemporal_store(f5,  fb + 2);
  }
  __builtin_nontemporal_store(f3,                   N_e   + e);
  __builtin_nontemporal_store((f5 - f2) * 0.5f,     M_mid + e);
  __builtin_nontemporal_store(f4,                   V_e   + e);
  __builtin_nontemporal_store(f2,                   M1_e  + e);
  __builtin_nontemporal_store(f5,                   M2_e  + e);
  __builtin_nontemporal_store(l0,                   l0_out + e);
  __builtin_nontemporal_store(c,                    c_out  + e);
  __builtin_nontemporal_store(s,                    s_out  + e);

  // --- scatter-add into L2-resident accumulator (global_atomic_add_f32) ----
  unsafeAtomicAdd(&nodal_forces[3 * nA + 0], FAx);
  unsafeAtomicAdd(&nodal_forces[3 * nA + 1], FAz);
  unsafeAtomicAdd(&nodal_forces[3 * nA + 2], f2);
  unsafeAtomicAdd(&nodal_forces[3 * nB + 0], FBx);
  unsafeAtomicAdd(&nodal_forces[3 * nB + 1], FBz);
  unsafeAtomicAdd(&nodal_forces[3 * nB + 2], f5);
}

extern "C" void kernel_launch(void* const* d_in, const int* in_sizes, int n_in,
                              void* d_out, int out_size, void* d_ws, size_t ws_size,
                              hipStream_t stream) {
  const float* pred_disp = (const float*)d_in[0];
  const int*   conn      = (const int*)  d_in[1];   // int32 pairs per harness contract
  const float* coords    = (const float*)d_in[2];
  const float* prop_E    = (const float*)d_in[3];
  const float* prop_A    = (const float*)d_in[4];
  const float* prop_I22  = (const float*)d_in[5];

  const int n_nodes = in_sizes[0] / 3;
  const int n_elems = in_sizes[3];

  float* out = (float*)d_out;
  long long off = 0;
  float* nodal   = out + off; off += 3ll * n_nodes;
  float* f_local = out + off; off += 6ll * n_elems;
  float* d_local = out + off; off += 6ll * n_elems;
  float* F_A     = out + off; off += 3ll * n_elems;
  float* F_B     = out + off; off += 3ll * n_elems;
  float* N_e     = out + off; off += n_elems;
  float* M_mid   = out + off; off += n_elems;
  float* V_e     = out + off; off += n_elems;
  float* M1_e    = out + off; off += n_elems;
  float* M2_e    = out + off; off += n_elems;
  float* l0_o    = out + off; off += n_elems;
  float* c_o     = out + off; off += n_elems;
  float* s_o     = out + off; off += n_elems;

  // Zero the atomic accumulator every call (harness replays without re-init).
  const long long nz = 3ll * n_nodes;
  beam2d_zero_kernel<<<(int)((nz + 255) / 256), 256, 0, stream>>>(nodal, nz);

  beam2d_elem_kernel<<<(n_elems + 255) / 256, 256, 0, stream>>>(
      pred_disp, conn, coords, prop_E, prop_A, prop_I22,
      nodal, f_local, d_local, F_A, F_B,
      N_e, M_mid, V_e, M1_e, M2_e, l0_o, c_o, s_o, n_elems);
}